// GIN_90117003805312
// MI455X (gfx1250) — compile-verified
//
#include <hip/hip_runtime.h>

#define N_NODES 100000
#define N_EDGES 1600000
#define D 128
#define NOUT 40
#define BM 32          // node rows per block
#define LDP 132        // padded LDS row pitch (floats) -> conflict-free A reads

typedef float v2f __attribute__((ext_vector_type(2)));
typedef float v8f __attribute__((ext_vector_type(8)));

// agg[i] = (1 + eps) * x[i]   (self term folded into scatter target)
__global__ void init_agg_kernel(const float* __restrict__ x,
                                const float* __restrict__ eps,
                                float* __restrict__ agg, int n4) {
  int i = blockIdx.x * blockDim.x + threadIdx.x;
  if (i >= n4) return;
  float s = 1.0f + eps[0];
  float4 v = ((const float4*)x)[i];
  v.x *= s; v.y *= s; v.z *= s; v.w *= s;
  ((float4*)agg)[i] = v;
}

// one wave (32 lanes) per edge; lane covers 4 consecutive floats of the row
__global__ void scatter_kernel(const float* __restrict__ feat,
                               const long long* __restrict__ src,
                               const long long* __restrict__ dst,
                               float* __restrict__ agg, int n_edges) {
  int gid  = blockIdx.x * blockDim.x + threadIdx.x;
  int e    = gid >> 5;
  int lane = gid & 31;
  if (e >= n_edges) return;
  long long s = src[e];
  long long d = dst[e];
  float4 v = *(const float4*)(feat + (size_t)s * D + lane * 4);
  float* p = agg + (size_t)d * D + lane * 4;
  unsafeAtomicAdd(p + 0, v.x);
  unsafeAtomicAdd(p + 1, v.y);
  unsafeAtomicAdd(p + 2, v.z);
  unsafeAtomicAdd(p + 3, v.w);
}

// Fused MLP: out = [relu?]( relu(in @ W1 + b1) @ W2 + b2 ), fp32 WMMA 16x16x4
__global__ __launch_bounds__(256)
void gin_mlp_kernel(const float* __restrict__ in,
                    const float* __restrict__ W1, const float* __restrict__ b1,
                    const float* __restrict__ W2, const float* __restrict__ b2,
                    float* __restrict__ out, int nout, int relu_out) {
  __shared__ float sIn[BM * LDP];
  __shared__ float sMid[BM * LDP];
  const int tid  = threadIdx.x;
  const int wave = tid >> 5;
  const int lane = tid & 31;
  const int l15  = lane & 15;
  const int hi   = lane >> 4;            // lane-half: K offset = 2*hi
  const int rowbase = blockIdx.x * BM;

  // cooperative stage of input tile (BM x 128), zero-padded out-of-range rows
  #pragma unroll
  for (int i = 0; i < (BM * (D / 4)) / 256; ++i) {   // 4 float4 per thread
    int idx = tid + i * 256;
    int r   = idx >> 5;                  // / (D/4)
    int c4  = idx & 31;
    float4 v = make_float4(0.f, 0.f, 0.f, 0.f);
    if (rowbase + r < N_NODES)
      v = *(const float4*)(in + (size_t)(rowbase + r) * D + c4 * 4);
    *(float4*)(sIn + r * LDP + c4 * 4) = v;
  }
  __syncthreads();

  // ---- GEMM1: (BM x 128) @ (128 x 128) + b1, ReLU -> sMid ----
  {
    const int c = wave;                  // col tile 0..7
    v8f acc0 = {}; v8f acc1 = {};
    const float* Wc = W1 + c * 16 + l15;
    #pragma unroll 8
    for (int k0 = 0; k0 < D; k0 += 4) {
      int kk = k0 + 2 * hi;
      v2f b;
      b.x = Wc[(size_t)kk * D];
      b.y = Wc[(size_t)(kk + 1) * D];
      v2f a0 = *(const v2f*)(sIn + (l15) * LDP + kk);        // rows 0..15
      v2f a1 = *(const v2f*)(sIn + (16 + l15) * LDP + kk);   // rows 16..31
      acc0 = __builtin_amdgcn_wmma_f32_16x16x4_f32(false, a0, false, b,
                                                   (short)0, acc0, false, false);
      acc1 = __builtin_amdgcn_wmma_f32_16x16x4_f32(false, a1, false, b,
                                                   (short)0, acc1, false, false);
    }
    float bias = b1[c * 16 + l15];
    #pragma unroll
    for (int j = 0; j < 8; ++j) {        // C layout: VGPR j -> rows j, j+8
      int rr = j + 8 * hi;
      sMid[rr * LDP + c * 16 + l15]        = fmaxf(acc0[j] + bias, 0.0f);
      sMid[(16 + rr) * LDP + c * 16 + l15] = fmaxf(acc1[j] + bias, 0.0f);
    }
  }
  __syncthreads();

  // ---- GEMM2: (BM x 128) @ (128 x nout) + b2, optional ReLU -> global ----
  const int ct = (nout + 15) / 16;
  for (int t = wave; t < 2 * ct; t += 8) {   // uniform per wave
    int r = t / ct, c = t % ct;
    int col = c * 16 + l15;
    bool colok = col < nout;
    v8f acc = {};
    #pragma unroll 8
    for (int k0 = 0; k0 < D; k0 += 4) {
      int kk = k0 + 2 * hi;
      v2f b;
      b.x = colok ? W2[(size_t)kk * nout + col] : 0.0f;
      b.y = colok ? W2[(size_t)(kk + 1) * nout + col] : 0.0f;
      v2f a = *(const v2f*)(sMid + (r * 16 + l15) * LDP + kk);
      acc = __builtin_amdgcn_wmma_f32_16x16x4_f32(false, a, false, b,
                                                  (short)0, acc, false, false);
    }
    float bias = colok ? b2[col] : 0.0f;
    #pragma unroll
    for (int j = 0; j < 8; ++j) {
      int row = rowbase + r * 16 + j + 8 * hi;
      float v = acc[j] + bias;
      if (relu_out) v = fmaxf(v, 0.0f);
      if (colok && row < N_NODES)
        out[(size_t)row * nout + col] = v;
    }
  }
}

// in-place log_softmax over rows of ncol (<=64); one wave per row
__global__ void log_softmax_kernel(float* __restrict__ o, int n, int ncol) {
  int gid  = blockIdx.x * blockDim.x + threadIdx.x;
  int row  = gid >> 5;
  int lane = gid & 31;
  if (row >= n) return;
  float* p = o + (size_t)row * ncol;
  const float NEG = -3.402823466e+38f;
  float v0 = (lane < ncol)      ? p[lane]      : NEG;
  float v1 = (lane + 32 < ncol) ? p[lane + 32] : NEG;
  float m = fmaxf(v0, v1);
  #pragma unroll
  for (int off = 16; off >= 1; off >>= 1) m = fmaxf(m, __shfl_xor(m, off, 32));
  float e = ((lane < ncol)      ? expf(v0 - m) : 0.0f) +
            ((lane + 32 < ncol) ? expf(v1 - m) : 0.0f);
  #pragma unroll
  for (int off = 16; off >= 1; off >>= 1) e += __shfl_xor(e, off, 32);
  float ls = logf(e);
  if (lane < ncol)      p[lane]      = v0 - m - ls;
  if (lane + 32 < ncol) p[lane + 32] = v1 - m - ls;
}

extern "C" void kernel_launch(void* const* d_in, const int* in_sizes, int n_in,
                              void* d_out, int out_size, void* d_ws, size_t ws_size,
                              hipStream_t stream) {
  const float*     x     = (const float*)d_in[0];
  const long long* edges = (const long long*)d_in[1];
  const long long* src   = edges;
  const long long* dstp  = edges + N_EDGES;
  const float* W1_0 = (const float*)d_in[2];  const float* b1_0 = (const float*)d_in[3];
  const float* W2_0 = (const float*)d_in[4];  const float* b2_0 = (const float*)d_in[5];
  const float* eps0 = (const float*)d_in[6];
  const float* W1_1 = (const float*)d_in[7];  const float* b1_1 = (const float*)d_in[8];
  const float* W2_1 = (const float*)d_in[9];  const float* b2_1 = (const float*)d_in[10];
  const float* eps1 = (const float*)d_in[11];
  const float* W1_2 = (const float*)d_in[12]; const float* b1_2 = (const float*)d_in[13];
  const float* W2_2 = (const float*)d_in[14]; const float* b2_2 = (const float*)d_in[15];
  const float* eps2 = (const float*)d_in[16];

  float* agg = (float*)d_ws;                      // 100000*128 f32
  float* h   = agg + (size_t)N_NODES * D;         // 100000*128 f32
  float* out = (float*)d_out;                     // 100000*40  f32

  dim3 blk(256);
  const int n4     = N_NODES * D / 4;
  const int g_elem = (n4 + 255) / 256;
  const int g_sc   = (int)(((size_t)N_EDGES * 32 + 255) / 256);
  const int g_gemm = (N_NODES + BM - 1) / BM;
  const int g_ls   = (N_NODES * 32 + 255) / 256;

  // layer 0
  init_agg_kernel<<<g_elem, blk, 0, stream>>>(x, eps0, agg, n4);
  scatter_kernel<<<g_sc, blk, 0, stream>>>(x, src, dstp, agg, N_EDGES);
  gin_mlp_kernel<<<g_gemm, blk, 0, stream>>>(agg, W1_0, b1_0, W2_0, b2_0, h, D, 1);
  // layer 1
  init_agg_kernel<<<g_elem, blk, 0, stream>>>(h, eps1, agg, n4);
  scatter_kernel<<<g_sc, blk, 0, stream>>>(h, src, dstp, agg, N_EDGES);
  gin_mlp_kernel<<<g_gemm, blk, 0, stream>>>(agg, W1_1, b1_1, W2_1, b2_1, h, D, 1);
  // layer 2 (no output ReLU), writes d_out
  init_agg_kernel<<<g_elem, blk, 0, stream>>>(h, eps2, agg, n4);
  scatter_kernel<<<g_sc, blk, 0, stream>>>(h, src, dstp, agg, N_EDGES);
  gin_mlp_kernel<<<g_gemm, blk, 0, stream>>>(agg, W1_2, b1_2, W2_2, b2_2, out, NOUT, 0);
  // log_softmax in place on d_out
  log_softmax_kernel<<<g_ls, blk, 0, stream>>>(out, N_NODES, NOUT);
}